// ProgressiveTSSA_Fusion1_11802570129756
// MI455X (gfx1250) — compile-verified
//
#include <hip/hip_runtime.h>
#include <math.h>

// MI455X / gfx1250, wave32. All matmuls use V_WMMA_F32_16X16X4_F32 (fp32 matrix
// path) so numerics match the fp32 reference. Workload is compute-bound
// (~40 GFLOP fp32 vs ~100MB traffic at 23.3 TB/s), dominated by the 3072-token
// MHA, handled with a flash-style streaming-softmax WMMA kernel.
// GEMM waves compute 16x32 output strips: each A-fragment feeds two WMMAs
// (operand reuse); the k-loop is pointer-incremented (no per-iter mul/select)
// and global_prefetch_b8 is issued one k-tile ahead.

typedef float v2f __attribute__((ext_vector_type(2)));
typedef float v8f __attribute__((ext_vector_type(8)));

#define CCH   128
#define HH    32
#define WW    32
#define HWSZ  1024
#define NTOK  1024
#define BB    2
#define T3    3072
#define HEADS 4
#define DHEAD 32

__device__ __forceinline__ float gelu_f(float x) {
  return 0.5f * x * (1.0f + erff(x * 0.70710678118654752440f));
}

// ---------------------------------------------------------------------------
// Generic batched GEMM:  O[b][m][n] = sum_k W[m*lda+k] * X(b)[k][n] + bias[m]
// X(b)[k][n] = (k<ksplit ? X0 : X1 shifted)[b*xbS + k*ldb + n]  (concat support)
// A-frag: lane m=lane&15, v0=K(koff), v1=K(koff+1), koff = lane<16?0:2
// B-frag: lane n=lane&15, same K striping. C/D: vgpr r -> row r (lanes<16), r+8.
// Each wave computes a 16x32 strip: one A-frag drives two WMMAs.
// ---------------------------------------------------------------------------
__global__ void __launch_bounds__(128) gemm_wmma_k(
    const float* __restrict__ W, int lda,
    const float* __restrict__ X0, const float* __restrict__ X1, int ksplit,
    int ldb, long xbS,
    const float* __restrict__ bias,
    float* __restrict__ O, int ldo, long obS,
    int M, int N, int K)
{
  int lane  = threadIdx.x & 31;
  int wv    = threadIdx.x >> 5;
  int tileM = (blockIdx.y * 4 + wv) * 16;
  if (tileM >= M) return;
  int tileN = blockIdx.x * 32;
  int b     = blockIdx.z;
  const float* x0 = X0 + (long)b * xbS;
  const float* x1 = X1 + (long)b * xbS;
  float*       o  = O  + (long)b * obS;

  int l16  = lane & 15;
  int koff = (lane < 16) ? 0 : 2;
  int mrow = tileM + l16;
  int n    = tileN + l16;

  v8f acc0 = {0.f,0.f,0.f,0.f,0.f,0.f,0.f,0.f};
  v8f acc1 = {0.f,0.f,0.f,0.f,0.f,0.f,0.f,0.f};

  const long xstep = 4L * ldb;
  const float* wp = W + (long)mrow * lda + koff;

  // segment 1: k in [0, min(ksplit,K))
  int kend0 = (ksplit < K) ? ksplit : K;
  const float* xs = x0 + (long)koff * ldb + n;
  for (int kk = 0; kk < kend0; kk += 4) {
    __builtin_prefetch(xs + xstep, 0, 0);
    v2f a;   a[0]   = wp[0];   a[1]   = wp[1];
    v2f bf0; bf0[0] = xs[0];   bf0[1] = xs[ldb];
    v2f bf1; bf1[0] = xs[16];  bf1[1] = xs[ldb + 16];
    acc0 = __builtin_amdgcn_wmma_f32_16x16x4_f32(false, a, false, bf0,
                                                 (short)0, acc0, false, false);
    acc1 = __builtin_amdgcn_wmma_f32_16x16x4_f32(false, a, false, bf1,
                                                 (short)0, acc1, false, false);
    wp += 4; xs += xstep;
  }
  // segment 2: k in [ksplit, K) sourced from X1 (concat input)
  if (ksplit < K) {
    const float* xt = x1 + (long)koff * ldb + n;
    for (int kk = ksplit; kk < K; kk += 4) {
      __builtin_prefetch(xt + xstep, 0, 0);
      v2f a;   a[0]   = wp[0];   a[1]   = wp[1];
      v2f bf0; bf0[0] = xt[0];   bf0[1] = xt[ldb];
      v2f bf1; bf1[0] = xt[16];  bf1[1] = xt[ldb + 16];
      acc0 = __builtin_amdgcn_wmma_f32_16x16x4_f32(false, a, false, bf0,
                                                   (short)0, acc0, false, false);
      acc1 = __builtin_amdgcn_wmma_f32_16x16x4_f32(false, a, false, bf1,
                                                   (short)0, acc1, false, false);
      wp += 4; xt += xstep;
    }
  }

  int rbase = tileM + ((lane < 16) ? 0 : 8);
#pragma unroll
  for (int r = 0; r < 8; ++r) {
    int row = rbase + r;
    float bv = bias ? bias[row] : 0.f;
    o[(long)row * ldo + n]      = acc0[r] + bv;
    o[(long)row * ldo + n + 16] = acc1[r] + bv;
  }
}

// ---------------------------------------------------------------------------
// Flash attention: one wave per (b, head, 16-query tile). qkv channel-major:
// qkv[b][row][tok], rows 0..127=Q, 128..255=K, 256..383=V; head h -> h*32+dd.
// ---------------------------------------------------------------------------
__global__ void __launch_bounds__(32) flash_attn_k(
    const float* __restrict__ qkv, float* __restrict__ attnO)
{
  int lane  = threadIdx.x & 31;
  int l16   = lane & 15;
  int koff  = (lane < 16) ? 0 : 2;
  int qbase = blockIdx.x * 16;
  int h     = blockIdx.y;
  int b     = blockIdx.z;
  const float* Q  = qkv + ((long)b * 384 + h * DHEAD) * T3;
  const float* Kp = Q + (long)128 * T3;
  const float* Vp = Q + (long)256 * T3;

  const float scl = 0.17677669529663687f;   // 1/sqrt(32)
  float qa0[8], qa1[8];
#pragma unroll
  for (int s = 0; s < 8; ++s) {
    int k0 = s * 4 + koff;
    qa0[s] = Q[(long)k0 * T3 + qbase + l16] * scl;
    qa1[s] = Q[(long)(k0 + 1) * T3 + qbase + l16] * scl;
  }
  v8f o0 = {0.f,0.f,0.f,0.f,0.f,0.f,0.f,0.f};
  v8f o1 = {0.f,0.f,0.f,0.f,0.f,0.f,0.f,0.f};
  float mrun[8], lrun[8];
#pragma unroll
  for (int r = 0; r < 8; ++r) { mrun[r] = -1e30f; lrun[r] = 0.f; }

  __shared__ float P[16][17];

  for (int kt = 0; kt < T3 / 16; ++kt) {
    int kb = kt * 16;
    // prefetch next key/value tiles (speculative; OOB dropped silently)
    __builtin_prefetch(Kp + (long)koff * T3 + kb + 16 + l16, 0, 0);
    __builtin_prefetch(Vp + (long)l16 * T3 + kb + 16 + koff, 0, 0);
    v8f s = {0.f,0.f,0.f,0.f,0.f,0.f,0.f,0.f};
#pragma unroll
    for (int st = 0; st < 8; ++st) {
      int k0 = st * 4 + koff;
      v2f a;  a[0] = qa0[st]; a[1] = qa1[st];
      v2f bf; bf[0] = Kp[(long)k0 * T3 + kb + l16];
              bf[1] = Kp[(long)(k0 + 1) * T3 + kb + l16];
      s = __builtin_amdgcn_wmma_f32_16x16x4_f32(false, a, false, bf,
                                                (short)0, s, false, false);
    }
    float p[8];
#pragma unroll
    for (int r = 0; r < 8; ++r) {
      float sv = s[r];
      float tm = sv;
      tm = fmaxf(tm, __shfl_xor(tm, 8, 16));
      tm = fmaxf(tm, __shfl_xor(tm, 4, 16));
      tm = fmaxf(tm, __shfl_xor(tm, 2, 16));
      tm = fmaxf(tm, __shfl_xor(tm, 1, 16));
      float mnew  = fmaxf(mrun[r], tm);
      float alpha = __expf(mrun[r] - mnew);
      float pv    = __expf(sv - mnew);
      float ps = pv;
      ps += __shfl_xor(ps, 8, 16);
      ps += __shfl_xor(ps, 4, 16);
      ps += __shfl_xor(ps, 2, 16);
      ps += __shfl_xor(ps, 1, 16);
      lrun[r] = lrun[r] * alpha + ps;
      mrun[r] = mnew;
      o0[r] *= alpha; o1[r] *= alpha;
      p[r] = pv;
    }
    __syncthreads();
    int prow = (lane < 16) ? 0 : 8;
#pragma unroll
    for (int r = 0; r < 8; ++r) P[prow + r][l16] = p[r];
    __syncthreads();
#pragma unroll
    for (int st = 0; st < 4; ++st) {
      int k0 = st * 4 + koff;
      v2f a;  a[0] = P[l16][k0]; a[1] = P[l16][k0 + 1];
      v2f b0; b0[0] = Vp[(long)l16 * T3 + kb + k0];
              b0[1] = Vp[(long)l16 * T3 + kb + k0 + 1];
      v2f b1; b1[0] = Vp[(long)(16 + l16) * T3 + kb + k0];
              b1[1] = Vp[(long)(16 + l16) * T3 + kb + k0 + 1];
      o0 = __builtin_amdgcn_wmma_f32_16x16x4_f32(false, a, false, b0,
                                                 (short)0, o0, false, false);
      o1 = __builtin_amdgcn_wmma_f32_16x16x4_f32(false, a, false, b1,
                                                 (short)0, o1, false, false);
    }
  }
  long obase = ((long)b * CCH + h * DHEAD) * T3;
  int half8 = (lane < 16) ? 0 : 8;
#pragma unroll
  for (int r = 0; r < 8; ++r) {
    float inv = 1.0f / lrun[r];
    int tok = qbase + r + half8;
    attnO[obase + (long)l16 * T3 + tok]        = o0[r] * inv;
    attnO[obase + (long)(16 + l16) * T3 + tok] = o1[r] * inv;
  }
}

// ----------------------- depthwise 3x3 (+bias +BN +GELU) --------------------
__global__ void dwconv3x3_k(const float* __restrict__ in, const float* __restrict__ w,
                            const float* __restrict__ bias,
                            const float* __restrict__ bn_m, const float* __restrict__ bn_v,
                            const float* __restrict__ bn_g, const float* __restrict__ bn_b,
                            float* __restrict__ out, int Cch, int total, int doGelu)
{
  int idx = blockIdx.x * blockDim.x + threadIdx.x;
  if (idx >= total) return;
  int x = idx % WW; int t = idx / WW;
  int y = t % HH;   t /= HH;
  int c = t % Cch;  int b = t / Cch;
  const float* ip = in + (long)(b * Cch + c) * HWSZ;
  const float* wp = w + c * 9;
  float acc = 0.f;
#pragma unroll
  for (int ky = 0; ky < 3; ++ky) {
    int iy = y + ky - 1;
    if ((unsigned)iy >= HH) continue;
#pragma unroll
    for (int kx = 0; kx < 3; ++kx) {
      int ix = x + kx - 1;
      if ((unsigned)ix >= WW) continue;
      acc += ip[iy * WW + ix] * wp[ky * 3 + kx];
    }
  }
  if (bias) acc += bias[c];
  if (bn_m) acc = (acc - bn_m[c]) * (bn_g[c] * rsqrtf(bn_v[c] + 1e-5f)) + bn_b[c];
  if (doGelu) acc = gelu_f(acc);
  out[idx] = acc;
}

// ------------- depthwise 7x7 + sm_b + residual, accumulated into out -------
__global__ void dw7add_k(const float* __restrict__ h, const float* __restrict__ w,
                         const float* __restrict__ smb, const float* __restrict__ cur,
                         float* __restrict__ out, int total)
{
  int idx = blockIdx.x * blockDim.x + threadIdx.x;
  if (idx >= total) return;
  int x = idx % WW; int t = idx / WW;
  int y = t % HH;   t /= HH;
  int c = t % CCH;  int b = t / CCH;
  const float* ip = h + (long)(b * CCH + c) * HWSZ;
  const float* wp = w + c * 49;
  float acc = 0.f;
  for (int ky = 0; ky < 7; ++ky) {
    int iy = y + ky - 3;
    if ((unsigned)iy >= HH) continue;
    for (int kx = 0; kx < 7; ++kx) {
      int ix = x + kx - 3;
      if ((unsigned)ix >= WW) continue;
      acc += ip[iy * WW + ix] * wp[ky * 7 + kx];
    }
  }
  out[idx] += acc + smb[c] + cur[idx];
}

__global__ void pff_final_k(const float* o0, const float* o1, const float* o2,
                            const float* sa, const float* xin, float* out, int total)
{
  int idx = blockIdx.x * blockDim.x + threadIdx.x;
  if (idx >= total) return;
  out[idx] = sa[0] * o0[idx] + sa[1] * o1[idx] + sa[2] * o2[idx] + xin[idx];
}

// ------------------------------ GroupNorm(1,C) -----------------------------
__global__ void gn_k(const float* __restrict__ in, const float* __restrict__ g,
                     const float* __restrict__ bt, float* __restrict__ out)
{
  int b = blockIdx.x, tid = threadIdx.x;
  const int total = CCH * HWSZ;
  const float* ip = in + (long)b * total;
  float* op = out + (long)b * total;
  __shared__ float r1[256], r2[256];
  float s = 0.f, sq = 0.f;
  for (int i = tid; i < total; i += 256) { float v = ip[i]; s += v; sq += v * v; }
  r1[tid] = s; r2[tid] = sq; __syncthreads();
  for (int st = 128; st > 0; st >>= 1) {
    if (tid < st) { r1[tid] += r1[tid + st]; r2[tid] += r2[tid + st]; }
    __syncthreads();
  }
  float mu  = r1[0] / (float)total;
  float var = r2[0] / (float)total - mu * mu;
  float rstd = rsqrtf(var + 1e-5f);
  for (int i = tid; i < total; i += 256) {
    int c = i >> 10;               // i / (H*W)
    op[i] = (ip[i] - mu) * rstd * g[c] + bt[c];
  }
}

// --------------------------- pooling + bilinear ----------------------------
__global__ void pool_k(const float* __restrict__ in, float* __restrict__ out,
                       int scale, int total)
{
  int idx = blockIdx.x * blockDim.x + threadIdx.x;
  if (idx >= total) return;
  int ph = HH / scale, pw = WW / scale;
  int px = idx % pw; int t = idx / pw;
  int py = t % ph;   t /= ph;
  int c = t % CCH;   int b = t / CCH;
  long base = ((long)(b * CCH + c) * HH + py * scale) * WW + px * scale;
  float s = 0.f;
  for (int dy = 0; dy < scale; ++dy)
    for (int dx = 0; dx < scale; ++dx)
      s += in[base + dy * WW + dx];
  out[idx] = s / (float)(scale * scale);
}

__global__ void upsample_k(const float* __restrict__ pooled, float* __restrict__ out,
                           int scale, int total)
{
  int idx = blockIdx.x * blockDim.x + threadIdx.x;
  if (idx >= total) return;
  int ph = HH / scale, pw = WW / scale;
  int x = idx % WW; int t = idx / WW;
  int y = t % HH;   t /= HH;
  int c = t % CCH;  int b = t / CCH;
  float fy = (y + 0.5f) / (float)scale - 0.5f;
  float fx = (x + 0.5f) / (float)scale - 0.5f;
  int y0 = (int)floorf(fy); float wy = fy - (float)y0;
  int x0 = (int)floorf(fx); float wx = fx - (float)x0;
  int y0c = min(max(y0, 0), ph - 1), y1c = min(max(y0 + 1, 0), ph - 1);
  int x0c = min(max(x0, 0), pw - 1), x1c = min(max(x0 + 1, 0), pw - 1);
  const float* pb = pooled + (long)(b * CCH + c) * ph * pw;
  float v00 = pb[y0c * pw + x0c], v01 = pb[y0c * pw + x1c];
  float v10 = pb[y1c * pw + x0c], v11 = pb[y1c * pw + x1c];
  out[idx] = (1.f - wy) * ((1.f - wx) * v00 + wx * v01)
           + wy * ((1.f - wx) * v10 + wx * v11);
}

// ----------------------- TSSA per-scale (one block per b,h) ----------------
__global__ void tssa_k(const float* __restrict__ qkv, const float* __restrict__ temps,
                       int scaleIdx, float* __restrict__ featsCat)
{
  int b = blockIdx.x / HEADS;
  int h = blockIdx.x % HEADS;
  int tid = threadIdx.x;
  __shared__ float pi[NTOK];
  __shared__ float red[256];
  __shared__ float attn_s[DHEAD];
  const float* qb = qkv + ((long)b * 384 + h * DHEAD) * NTOK;
  const float* kb = qkv + ((long)b * 384 + 128 + h * DHEAD) * NTOK;
  const float* vb = qkv + ((long)b * 384 + 256 + h * DHEAD) * NTOK;
  float temp = temps[scaleIdx * HEADS + h];

  float lmax = -1e30f;
  for (int t = tid; t < NTOK; t += 256) {
    float ss = 0.f;
    for (int dd = 0; dd < DHEAD; ++dd) { float q = qb[dd * NTOK + t]; ss += q * q; }
    float nrm = sqrtf(ss);
    float dn = fmaxf(nrm, 1e-12f);
    float lg = (ss / (dn * dn)) * temp;
    pi[t] = lg;
    lmax = fmaxf(lmax, lg);
  }
  red[tid] = lmax; __syncthreads();
  for (int st = 128; st > 0; st >>= 1) {
    if (tid < st) red[tid] = fmaxf(red[tid], red[tid + st]);
    __syncthreads();
  }
  lmax = red[0]; __syncthreads();
  float lsum = 0.f;
  for (int t = tid; t < NTOK; t += 256) { float e = __expf(pi[t] - lmax); pi[t] = e; lsum += e; }
  red[tid] = lsum; __syncthreads();
  for (int st = 128; st > 0; st >>= 1) {
    if (tid < st) red[tid] += red[tid + st];
    __syncthreads();
  }
  float inv = 1.0f / red[0]; __syncthreads();
  for (int t = tid; t < NTOK; t += 256) pi[t] *= inv;
  __syncthreads();

  int dd = tid & 31, sl = tid >> 5;
  float part = 0.f;
  for (int t = sl * 128; t < sl * 128 + 128; ++t) {
    float kv = kb[dd * NTOK + t];
    part += pi[t] * kv * kv;
  }
  red[tid] = part; __syncthreads();
  if (tid < DHEAD) {
    float d = 0.f;
    for (int s = 0; s < 8; ++s) d += red[s * 32 + tid];
    attn_s[tid] = 1.0f / (1.0f + d);
  }
  __syncthreads();

  long base = ((long)b * CCH + h * DHEAD) * T3 + (long)scaleIdx * NTOK;
  for (int idx = tid; idx < DHEAD * NTOK; idx += 256) {
    int d2 = idx >> 10, t = idx & 1023;
    featsCat[base + (long)d2 * T3 + t] = -(vb[d2 * NTOK + t] * pi[t]) * attn_s[d2];
  }
}

// ------------------------------- small elementwise -------------------------
__global__ void mean3_k(const float* __restrict__ fp, float* __restrict__ fm, int total)
{
  int idx = blockIdx.x * blockDim.x + threadIdx.x;
  if (idx >= total) return;
  int t = idx % NTOK; int r = idx / NTOK;
  int c = r % CCH; int b = r / CCH;
  long fb = ((long)b * CCH + c) * T3 + t;
  fm[idx] = (fp[fb] + fp[fb + NTOK] + fp[fb + 2 * NTOK]) * (1.0f / 3.0f);
}

__global__ void resadd_k(const float* __restrict__ X, const float* __restrict__ a,
                         const float* __restrict__ rw, float* __restrict__ out, int total)
{
  int idx = blockIdx.x * blockDim.x + threadIdx.x;
  if (idx >= total) return;
  out[idx] = X[idx] + rw[0] * a[idx];
}

__global__ void gate_k(const float* __restrict__ dwh, float* __restrict__ gated, int total)
{
  int idx = blockIdx.x * blockDim.x + threadIdx.x;
  if (idx >= total) return;
  int t = idx % NTOK; int r = idx / NTOK;
  int c = r % 512; int b = r / 512;
  long ib = (long)b * 1024 * NTOK;
  float a = dwh[ib + (long)c * NTOK + t];
  float g = dwh[ib + (long)(c + 512) * NTOK + t];
  gated[idx] = gelu_f(a) * g;
}

// ------------- per-patch rfft2 * filter * irfft2 + final residual ----------
__global__ void __launch_bounds__(64) fft_patch_k(
    const float* __restrict__ hin, const float* __restrict__ fftw,
    const float* __restrict__ x1, const float* __restrict__ rw2,
    float* __restrict__ out)
{
  int tid = threadIdx.x;
  int pidx = blockIdx.x;             // 16 patches per plane
  int c = blockIdx.y, b = blockIdx.z;
  int py = pidx >> 2, px = pidx & 3;
  __shared__ float patch[64];
  __shared__ float Gre[40], Gim[40];
  long base = ((long)(b * CCH + c) * HH + py * 8) * WW + px * 8;
  int y = tid >> 3, x = tid & 7;
  patch[tid] = hin[base + y * WW + x];
  __syncthreads();
  if (tid < 40) {
    int u = tid / 5, v = tid % 5;
    float gr = 0.f, gi = 0.f;
    for (int yy = 0; yy < 8; ++yy)
      for (int xx = 0; xx < 8; ++xx) {
        float ang = -0.78539816339744830962f * (float)(u * yy + v * xx); // -2pi/8
        float sn, cs; __sincosf(ang, &sn, &cs);
        float pv = patch[yy * 8 + xx];
        gr += pv * cs; gi += pv * sn;
      }
    float f = fftw[((long)c * 8 + u) * 5 + v];
    Gre[tid] = gr * f; Gim[tid] = gi * f;
  }
  __syncthreads();
  float acc = 0.f;
  for (int fq = 0; fq < 40; ++fq) {
    int u = fq / 5, v = fq % 5;
    float w = (v == 0 || v == 4) ? 1.0f : 2.0f;   // Hermitian column weights
    float ang = 0.78539816339744830962f * (float)(u * y + v * x);
    float sn, cs; __sincosf(ang, &sn, &cs);
    acc += w * (Gre[fq] * cs - Gim[fq] * sn);
  }
  acc *= (1.0f / 64.0f);
  long oi = base + y * WW + x;
  out[oi] = x1[oi] + rw2[0] * acc;
}

// ===========================================================================
extern "C" void kernel_launch(void* const* d_in, const int* in_sizes, int n_in,
                              void* d_out, int out_size, void* d_ws, size_t ws_size,
                              hipStream_t stream)
{
  (void)in_sizes; (void)n_in; (void)out_size; (void)ws_size;
  auto F = [&](int i) -> const float* { return (const float*)d_in[i]; };

  float* Wp = (float*)d_ws;
  size_t off = 0;
  auto alloc = [&](size_t n) -> float* { float* p = Wp + off; off += n; return p; };
  const size_t PLANE = (size_t)BB * CCH * NTOK;          // 262144 floats
  float* curA     = alloc(PLANE);
  float* curBf    = alloc(PLANE);
  float* hb       = alloc(PLANE);
  float* ob0      = alloc(PLANE);
  float* ob1      = alloc(PLANE);
  float* ob2      = alloc(PLANE);
  float* pres     = alloc(PLANE);
  float* xgn      = alloc(PLANE);
  float* pooled   = alloc((size_t)BB * CCH * 256);
  float* xsb      = alloc(PLANE);
  float* qkvs     = alloc((size_t)BB * 384 * NTOK);
  float* featsCat = alloc((size_t)BB * CCH * T3);
  float* qkv2     = alloc((size_t)BB * 384 * T3);
  float* attnO    = alloc((size_t)BB * CCH * T3);
  float* fproj    = alloc((size_t)BB * CCH * T3);
  float* fmean    = alloc(PLANE);
  float* abuf     = alloc(PLANE);
  float* x1       = alloc(PLANE);
  float* h2       = alloc(PLANE);
  float* h1e      = alloc((size_t)BB * 1024 * NTOK);
  float* dwh      = alloc((size_t)BB * 1024 * NTOK);
  float* gated    = alloc((size_t)BB * 512 * NTOK);
  float* hout     = alloc(PLANE);

  auto gemm = [&](const float* Wm, int lda, const float* X0, const float* X1,
                  int ksplit, int ldb, long xbS, const float* bias,
                  float* O, int ldo, long obS, int M, int N, int K) {
    dim3 g((unsigned)(N / 32), (unsigned)((M + 63) / 64), (unsigned)BB);
    gemm_wmma_k<<<g, 128, 0, stream>>>(Wm, lda, X0, X1, ksplit, ldb, xbS,
                                       bias, O, ldo, obS, M, N, K);
  };
  auto blocks = [&](int total) { return dim3((unsigned)((total + 255) / 256)); };

  // params: 0=x, pff1 base=1, pff2 base=36, 71/72=gn, 73..80=attn,
  // 81..84=ffn, 85=res_w1, 86=res_w2
  auto run_pff = [&](int base, const float* xin, float* result) {
    const float* cur = xin;
    float* curbufs[2] = { curA, curBf };
    float* outs[3]    = { ob0, ob1, ob2 };
    int total = (int)PLANE;
    for (int i = 0; i < 3; ++i) {
      const float* dw_w = F(base + i * 10 + 0);
      const float* dw_b = F(base + i * 10 + 1);
      const float* bn_g = F(base + i * 10 + 2);
      const float* bn_b = F(base + i * 10 + 3);
      const float* bn_m = F(base + i * 10 + 4);
      const float* bn_v = F(base + i * 10 + 5);
      const float* cm_w = F(base + i * 10 + 6);
      const float* cm_b = F(base + i * 10 + 7);
      const float* sm_w = F(base + i * 10 + 8);
      const float* sm_b = F(base + i * 10 + 9);
      dwconv3x3_k<<<blocks(total), 256, 0, stream>>>(
          cur, dw_w, dw_b, bn_m, bn_v, bn_g, bn_b, hb, CCH, total, 1);
      gemm(cm_w, 128, hb, hb, 128, HWSZ, (long)CCH * HWSZ, cm_b,
           outs[i], HWSZ, (long)CCH * HWSZ, 128, HWSZ, 128);
      dw7add_k<<<blocks(total), 256, 0, stream>>>(hb, sm_w, sm_b, cur, outs[i], total);
      if (i < 2) {
        const float* fw = F(base + 30 + i);
        const float* fb = F(base + 32 + i);
        gemm(fw, 256, cur, outs[i], 128, HWSZ, (long)CCH * HWSZ, fb,
             curbufs[i], HWSZ, (long)CCH * HWSZ, 128, HWSZ, 256);
        cur = curbufs[i];
      }
    }
    pff_final_k<<<blocks(total), 256, 0, stream>>>(ob0, ob1, ob2, F(base + 34),
                                                   xin, result, total);
  };

  const float* X = F(0);
  run_pff(1, X, pres);
  gn_k<<<BB, 256, 0, stream>>>(pres, F(71), F(72), xgn);

  const int scales[3] = {1, 2, 4};
  for (int i = 0; i < 3; ++i) {
    int sc = scales[i];
    const float* xsp = xgn;
    if (sc > 1) {
      int pt = BB * CCH * (HH / sc) * (WW / sc);
      pool_k<<<blocks(pt), 256, 0, stream>>>(xgn, pooled, sc, pt);
      int ut = (int)PLANE;
      upsample_k<<<blocks(ut), 256, 0, stream>>>(pooled, xsb, sc, ut);
      xsp = xsb;
    }
    gemm(F(73) + (size_t)i * 384 * 128, 128, xsp, xsp, 128, NTOK,
         (long)CCH * NTOK, nullptr, qkvs, NTOK, (long)384 * NTOK, 384, NTOK, 128);
    tssa_k<<<BB * HEADS, 256, 0, stream>>>(qkvs, F(74), i, featsCat);
  }

  gemm(F(75), 128, featsCat, featsCat, 128, T3, (long)CCH * T3, F(76),
       qkv2, T3, (long)384 * T3, 384, T3, 128);
  dim3 fg((unsigned)(T3 / 16), HEADS, BB);
  flash_attn_k<<<fg, 32, 0, stream>>>(qkv2, attnO);
  gemm(F(77), 128, attnO, attnO, 128, T3, (long)CCH * T3, F(78),
       fproj, T3, (long)CCH * T3, 128, T3, 128);
  mean3_k<<<blocks((int)PLANE), 256, 0, stream>>>(fproj, fmean, (int)PLANE);
  gemm(F(79), 128, fmean, fmean, 128, NTOK, (long)CCH * NTOK, F(80),
       abuf, NTOK, (long)CCH * NTOK, 128, NTOK, 128);
  resadd_k<<<blocks((int)PLANE), 256, 0, stream>>>(X, abuf, F(85), x1, (int)PLANE);

  run_pff(36, x1, h2);

  gemm(F(81), 128, h2, h2, 128, NTOK, (long)CCH * NTOK, nullptr,
       h1e, NTOK, (long)1024 * NTOK, 1024, NTOK, 128);
  int te = BB * 1024 * HWSZ;
  dwconv3x3_k<<<blocks(te), 256, 0, stream>>>(h1e, F(82), nullptr, nullptr,
                                              nullptr, nullptr, nullptr,
                                              dwh, 1024, te, 0);
  int tg = BB * 512 * NTOK;
  gate_k<<<blocks(tg), 256, 0, stream>>>(dwh, gated, tg);
  gemm(F(83), 512, gated, gated, 512, NTOK, (long)512 * NTOK, nullptr,
       hout, NTOK, (long)CCH * NTOK, 128, NTOK, 512);

  dim3 fftg(16, CCH, BB);
  fft_patch_k<<<fftg, 64, 0, stream>>>(hout, F(84), x1, F(86), (float*)d_out);
}